// LiMNet_12584254177655
// MI455X (gfx1250) — compile-verified
//
#include <hip/hip_runtime.h>
#include <math.h>

// LiMNet edge-GRU update for MI455X (gfx1250, wave32).
//
// h0 == 0  =>  gh == bias_hh, weight_hh is dead, GRU collapses to:
//   gi = X @ W_ih^T ;  r=sig(gi_r+bih_r+bhh_r); z=sig(gi_z+bih_z+bhh_z)
//   n=tanh(gi_n+bih_n+r*bhh_n); out=(1-z)*n
// item rows reuse the same GEMM with swapped X halves -> single logical GEMM
// M=2B=32768, K=256, N=384 (6.4 GFLOP).  Dominant cost: streaming the two
// 256 MB memory tensors into d_out (~1 GB HBM traffic, ~45us @ 23.3 TB/s),
// so kernel 1 is a nontemporal b128 copy and kernel 2 does gather + WMMA
// GEMM (f32 16x16x4, matching reference precision) + GRU epilogue + scatter.

typedef __attribute__((ext_vector_type(2))) float v2f;
typedef __attribute__((ext_vector_type(4))) float v4f;
typedef __attribute__((ext_vector_type(8))) float v8f;

static constexpr int kE  = 128;   // embedding size
static constexpr int k2E = 256;   // GEMM K
static constexpr int k3E = 384;   // GEMM N

__device__ __forceinline__ float sigmoid_f(float x) {
    return 1.0f / (1.0f + __expf(-x));
}

// ---------------------------------------------------------------------------
// Kernel 1: stream both memories into the output (nontemporal: 512 MB > L2).
// ---------------------------------------------------------------------------
__global__ void __launch_bounds__(256)
limnet_copy_kernel(const v4f* __restrict__ um, const v4f* __restrict__ im,
                   v4f* __restrict__ out_um, v4f* __restrict__ out_im,
                   long n_user4, long n_item4) {
    const long stride = (long)gridDim.x * blockDim.x;
    long i = (long)blockIdx.x * blockDim.x + threadIdx.x;
    for (long t = i; t < n_user4; t += stride) {
        v4f v = __builtin_nontemporal_load(um + t);
        __builtin_nontemporal_store(v, out_um + t);
    }
    for (long t = i; t < n_item4; t += stride) {
        v4f v = __builtin_nontemporal_load(im + t);
        __builtin_nontemporal_store(v, out_im + t);
    }
}

// ---------------------------------------------------------------------------
// Kernel 2: gather -> WMMA f32 GEMM -> fused GRU epilogue -> scatter.
// Block: 256 threads = 8 wave32.  16 edges/block -> 32x384 output tile.
// Waves tiled 2(M) x 4(N-groups); each wave owns 6 16x16 N-tiles.
// ---------------------------------------------------------------------------
__global__ void __launch_bounds__(256)
limnet_gru_kernel(const int* __restrict__ users, const int* __restrict__ items,
                  const float* __restrict__ user_mem, const float* __restrict__ item_mem,
                  const float* __restrict__ Wih,
                  const float* __restrict__ bias_ih, const float* __restrict__ bias_hh,
                  float* __restrict__ out_user_emb, float* __restrict__ out_item_emb,
                  float* __restrict__ out_user_mem, float* __restrict__ out_item_mem,
                  int nEdges) {
    // Padded LDS strides: 264 = 256+8 and 388 = 384+4 break the 64-bank
    // row-aliasing (256 % 64 == 0, 384 % 64 == 0 would serialize).
    constexpr int XS = 264;  // X tile row stride (floats), 16B multiple
    constexpr int GS = 388;  // gi tile row stride (floats)
    __shared__ float xs[32 * XS];  // rows 0..15 = [u|i], rows 16..31 = [i|u]
    __shared__ float gs[32 * GS];  // gi accumulator staging for epilogue

    const int tid   = threadIdx.x;
    const int edge0 = blockIdx.x * 16;

    // ---- Stage gathered X tile (each 128-float vector written to 2 spots) --
    for (int t = tid; t < 16 * 64; t += 256) {
        const int r = t >> 6;        // local edge 0..15
        const int q = t & 63;        // float4 slot within the 2E row
        int edge = edge0 + r;
        if (edge >= nEdges) edge = nEdges - 1;  // tail clamp (B % 16 == 0 anyway)
        if (q < 32) {
            const int u = users[edge];
            const v4f v = *(const v4f*)(user_mem + (size_t)u * kE + q * 4);
            *(v4f*)(xs + r * XS + q * 4)              = v;  // user row, left half
            *(v4f*)(xs + (16 + r) * XS + kE + q * 4)  = v;  // item row, right half
        } else {
            const int it = items[edge];
            const int qq = q - 32;
            const v4f v = *(const v4f*)(item_mem + (size_t)it * kE + qq * 4);
            *(v4f*)(xs + r * XS + kE + qq * 4)        = v;  // user row, right half
            *(v4f*)(xs + (16 + r) * XS + qq * 4)      = v;  // item row, left half
        }
    }
    __syncthreads();

    // ---- WMMA f32 16x16x4 main loop -----------------------------------
    const int lane   = tid & 31;
    const int wave   = tid >> 5;
    const int lane15 = lane & 15;
    const int h2     = (lane >> 4) << 1;   // hi half-lanes own K+2/K+3
    const int m0     = (wave >> 2) * 16;   // 0 or 16 (user vs item rows)
    const int g      = wave & 3;           // N-group: 96 columns each

    // B fragment base pointers: B[k][n] = Wih[n][k] (row-major [384,256]),
    // per-lane float2 at (n0+lane15)*256 + h2 + k  ->  (v0,v1)=(K even,K odd).
    const float* wp[6];
#pragma unroll
    for (int j = 0; j < 6; ++j) {
        const int n = g * 96 + j * 16 + lane15;
        wp[j] = Wih + (size_t)n * k2E + h2;
        __builtin_prefetch(wp[j], 0, 1);   // global_prefetch_b8 into GL2
    }
    const float* xp = xs + (m0 + lane15) * XS + h2;  // A fragment base (LDS)

    v8f acc[6] = {};
#pragma unroll 4
    for (int k = 0; k < k2E; k += 4) {
        const v2f a = *(const v2f*)(xp + k);
#pragma unroll
        for (int j = 0; j < 6; ++j) {
            const v2f b = *(const v2f*)(wp[j] + k);
            acc[j] = __builtin_amdgcn_wmma_f32_16x16x4_f32(
                false, a, false, b, (short)0, acc[j], false, false);
        }
    }

    // ---- Spill accumulators to LDS in C/D layout -----------------------
    // VGPR v, lane: row = m0 + v + 8*(lane>>4), col = n0 + (lane&15)
    const int hi8 = (lane >> 4) * 8;
#pragma unroll
    for (int j = 0; j < 6; ++j) {
        const int n0 = g * 96 + j * 16;
        float* gp = gs + (m0 + hi8) * GS + n0 + lane15;
#pragma unroll
        for (int v = 0; v < 8; ++v)
            gp[v * GS] = acc[j][v];
    }
    __syncthreads();

    // ---- Fused GRU epilogue + dense write + scatter --------------------
    for (int t = tid; t < 32 * kE; t += 256) {
        const int row = t >> 7;    // 0..31
        const int e   = t & 127;
        const float gr = gs[row * GS + e]            + bias_ih[e];
        const float gz = gs[row * GS + e + kE]       + bias_ih[e + kE];
        const float gn = gs[row * GS + e + 2 * kE]   + bias_ih[e + 2 * kE];
        const float r  = sigmoid_f(gr + bias_hh[e]);
        const float z  = sigmoid_f(gz + bias_hh[e + kE]);
        const float n  = tanhf(gn + r * bias_hh[e + 2 * kE]);
        const float o  = (1.0f - z) * n;   // + z*h with h==0

        int edge = edge0 + (row & 15);
        if (edge >= nEdges) continue;
        if (row < 16) {
            out_user_emb[(size_t)edge * kE + e] = o;
            out_user_mem[(size_t)users[edge] * kE + e] = o;
        } else {
            out_item_emb[(size_t)edge * kE + e] = o;
            out_item_mem[(size_t)items[edge] * kE + e] = o;
        }
    }
}

// ---------------------------------------------------------------------------
extern "C" void kernel_launch(void* const* d_in, const int* in_sizes, int n_in,
                              void* d_out, int out_size, void* d_ws, size_t ws_size,
                              hipStream_t stream) {
    const int*   users    = (const int*)d_in[0];
    const int*   items    = (const int*)d_in[1];
    const float* user_mem = (const float*)d_in[2];
    const float* item_mem = (const float*)d_in[3];
    const float* Wih      = (const float*)d_in[4];
    // d_in[5] (weight_hh) is mathematically dead: h0 == 0.
    const float* bias_ih  = (const float*)d_in[6];
    const float* bias_hh  = (const float*)d_in[7];

    const int  B       = in_sizes[0];
    const long nUserF  = in_sizes[2];   // NB_USERS * 128
    const long nItemF  = in_sizes[3];   // NB_ITEMS * 128

    float* out          = (float*)d_out;
    float* out_user_emb = out;
    float* out_item_emb = out_user_emb + (size_t)B * kE;
    float* out_user_mem = out_item_emb + (size_t)B * kE;
    float* out_item_mem = out_user_mem + (size_t)nUserF;

    // 1) Bandwidth phase: stream both memories into d_out (nontemporal).
    limnet_copy_kernel<<<4096, 256, 0, stream>>>(
        (const v4f*)user_mem, (const v4f*)item_mem,
        (v4f*)out_user_mem, (v4f*)out_item_mem, nUserF / 4, nItemF / 4);

    // 2) Compute phase (stream-ordered after the copy so scatter wins).
    limnet_gru_kernel<<<(B + 15) / 16, 256, 0, stream>>>(
        users, items, user_mem, item_mem, Wih, bias_ih, bias_hh,
        out_user_emb, out_item_emb, out_user_mem, out_item_mem, B);
}